// Model_40827959115956
// MI455X (gfx1250) — compile-verified
//
#include <hip/hip_runtime.h>
#include <cstddef>
#include <initializer_list>

// ---------------------------------------------------------------------------
// GRU stack for MI455X (gfx1250, wave32).
// Latency-bound recurrent scan: per direction-layer, one wave per 16-row batch
// tile runs the full T=2048 scan. h@Whh^T and x_t@Wih^T use
// V_WMMA_F32_16X16X4_F32 (fp32 recurrence, matching the reference numerics);
// gate nonlinearities use the gfx1250 V_TANH_F32 path; x-tiles are
// double-buffered in registers so global-load latency overlaps compute; h is
// bounced through LDS each step to convert WMMA C-layout -> next A-layout.
// Up to 4 independent direction-scans are batched per launch (fwd/rev x both
// branches), which is the maximum concurrency the dependence graph allows.
// ---------------------------------------------------------------------------

typedef float v2f __attribute__((ext_vector_type(2)));
typedef float v8f __attribute__((ext_vector_type(8)));

#define GRU_T 2048
#define GRU_B 128
#define GRU_H 16

struct ScanJob {
  const float* in;    // input sequence base
  const float* wih;   // (48, I) row-major
  const float* whh;   // (48, 16) row-major
  const float* bih;   // (48) or null
  const float* bhh;   // (48) or null
  const float* h0;    // (B, 16) or null (zeros)
  float* out;         // output base
  float* hT;          // (B, 16) final hidden or null
  int I, its, ibs;    // input width, t-stride, b-stride (elements)
  int ots, obs, ooff; // output t-stride, b-stride, channel offset
  int rev;            // 1 = reverse-time scan
};

struct JobPack { ScanJob j[4]; };

__device__ __forceinline__ float fast_tanh(float x) {
#if __has_builtin(__builtin_amdgcn_tanhf)
  return __builtin_amdgcn_tanhf(x);          // v_tanh_f32
#else
  float e = __expf(-2.0f * x);               // v_exp_f32 (base-e via ocml fast)
  return 2.0f * __builtin_amdgcn_rcpf(1.0f + e) - 1.0f;  // v_rcp_f32, no IEEE div
#endif
}
__device__ __forceinline__ float fast_sigm(float x) {
#if __has_builtin(__builtin_amdgcn_tanhf)
  return 0.5f + 0.5f * __builtin_amdgcn_tanhf(0.5f * x);
#else
  return __builtin_amdgcn_rcpf(1.0f + __expf(-x));
#endif
}

__global__ __launch_bounds__(32) void gru_scan(JobPack jp) {
  const ScanJob jb = jp.j[blockIdx.y];
  const int lane = threadIdx.x;        // 0..31, one wave per block
  const int half = lane >> 4;          // 0 | 1
  const int l16  = lane & 15;
  const int row0 = blockIdx.x * 16;    // batch tile base row

  __shared__ float hsh[GRU_H * GRU_H]; // h tile, [m][k] row-major

  // ---- Whh into WMMA B-layout registers: wh[gate][kchunk] ----
  // B(k,n) = Whh[n][k]; lane pair (l16, half): VGPR0 K=4c+2*half, VGPR1 K+1.
  v2f wh[3][4];
#pragma unroll
  for (int g = 0; g < 3; ++g)
#pragma unroll
    for (int c = 0; c < 4; ++c) {
      int n = g * 16 + l16;
      int k = c * 4 + half * 2;
      v2f w; w.x = jb.whh[n * 16 + k]; w.y = jb.whh[n * 16 + k + 1];
      wh[g][c] = w;
    }

  // ---- Wih into B-layout registers, zero-padded past I ----
  v2f wi[3][8];
#pragma unroll
  for (int g = 0; g < 3; ++g)
#pragma unroll
    for (int c = 0; c < 8; ++c) {
      int n = g * 16 + l16;
      int k = c * 4 + half * 2;
      int i0 = (k     < jb.I) ? k     : 0;
      int i1 = (k + 1 < jb.I) ? k + 1 : 0;
      float w0 = jb.wih[n * jb.I + i0];
      float w1 = jb.wih[n * jb.I + i1];
      v2f w;
      w.x = (k     < jb.I) ? w0 : 0.0f;
      w.y = (k + 1 < jb.I) ? w1 : 0.0f;
      wi[g][c] = w;
    }

  // ---- biases broadcast along N (lane) ----
  float bi[3], bh[3];
#pragma unroll
  for (int g = 0; g < 3; ++g) {
    bi[g] = jb.bih ? jb.bih[g * 16 + l16] : 0.0f;
    bh[g] = jb.bhh ? jb.bhh[g * 16 + l16] : 0.0f;
  }

  // ---- initial hidden in C-layout: hc[v] = h(m = v + 8*half, n = l16) ----
  v8f hc;
#pragma unroll
  for (int v = 0; v < 8; ++v) {
    int m = half * 8 + v;
    hc[v] = jb.h0 ? jb.h0[(size_t)(row0 + m) * GRU_H + l16] : 0.0f;
  }
#pragma unroll
  for (int v = 0; v < 8; ++v) {
    int m = half * 8 + v;
    hsh[m * 16 + l16] = hc[v];
  }
  __syncthreads(); // single-wave workgroup: cheap; orders DS before loop

  const int kc = (jb.I + 3) >> 2; // uniform number of valid k-chunks (4/7/8)

  // ---- incrementally updated element offsets (no per-step s_mul chains) ----
  const int t0  = jb.rev ? (GRU_T - 1) : 0;
  const int dxi = jb.rev ? -jb.its : jb.its;
  const int doi = jb.rev ? -jb.ots : jb.ots;
  int xoff = t0 * jb.its + (row0 + l16) * jb.ibs;
  int ooff = t0 * jb.ots + (row0 + half * 8) * jb.obs + jb.ooff + l16;

  // A-tile loader: clamped-index loads + cndmask zeroing (no EXEC branches)
  auto load_xtile = [&](int off, v2f (&dst)[8]) {
#pragma unroll
    for (int c = 0; c < 8; ++c) {
      if (c < kc) { // uniform guard
        const int k  = c * 4 + half * 2;
        const int i0 = (k     < jb.I) ? k     : 0;
        const int i1 = (k + 1 < jb.I) ? k + 1 : 0;
        float v0 = jb.in[off + i0];
        float v1 = jb.in[off + i1];
        dst[c].x = (k     < jb.I) ? v0 : 0.0f;
        dst[c].y = (k + 1 < jb.I) ? v1 : 0.0f;
      } else {
        dst[c].x = 0.0f; dst[c].y = 0.0f;
      }
    }
  };

  v2f xa[8];
  load_xtile(xoff, xa); // prologue: tile for t0

  for (int step = 0; step < GRU_T; ++step) {
    // ---- issue next step's x-tile loads first (latency hidden) ----
    const int xoff_n = (step + 1 < GRU_T) ? (xoff + dxi) : xoff;
    v2f xn[8];
    load_xtile(xoff_n, xn);
    if (step + 2 < GRU_T)
      __builtin_prefetch(jb.in + xoff + 2 * dxi, 0, 1); // global_prefetch

    // ---- recurrent gates: hg = h @ Whh^T + bhh (critical path) ----
    v8f hg0, hg1, hg2;
#pragma unroll
    for (int v = 0; v < 8; ++v) { hg0[v] = bh[0]; hg1[v] = bh[1]; hg2[v] = bh[2]; }
#pragma unroll
    for (int c = 0; c < 4; ++c) {
      v2f a; // A-layout read of h from LDS: (m = l16, k = 4c + 2*half [+1])
      int kk = c * 4 + half * 2;
      a.x = hsh[l16 * 16 + kk];
      a.y = hsh[l16 * 16 + kk + 1];
      hg0 = __builtin_amdgcn_wmma_f32_16x16x4_f32(false, a, false, wh[0][c],
                                                  (short)0, hg0, false, false);
      hg1 = __builtin_amdgcn_wmma_f32_16x16x4_f32(false, a, false, wh[1][c],
                                                  (short)0, hg1, false, false);
      hg2 = __builtin_amdgcn_wmma_f32_16x16x4_f32(false, a, false, wh[2][c],
                                                  (short)0, hg2, false, false);
    }

    // ---- input gates: xg = x_t @ Wih^T + bih (independent of h) ----
    v8f xg0, xg1, xg2;
#pragma unroll
    for (int v = 0; v < 8; ++v) { xg0[v] = bi[0]; xg1[v] = bi[1]; xg2[v] = bi[2]; }
#pragma unroll
    for (int c = 0; c < 8; ++c) {
      if (c < kc) { // uniform: EXEC stays all-ones for WMMA
        xg0 = __builtin_amdgcn_wmma_f32_16x16x4_f32(false, xa[c], false, wi[0][c],
                                                    (short)0, xg0, false, false);
        xg1 = __builtin_amdgcn_wmma_f32_16x16x4_f32(false, xa[c], false, wi[1][c],
                                                    (short)0, xg1, false, false);
        xg2 = __builtin_amdgcn_wmma_f32_16x16x4_f32(false, xa[c], false, wi[2][c],
                                                    (short)0, xg2, false, false);
      }
    }

    // ---- gate nonlinearities + state update (PyTorch GRU cell) ----
#pragma unroll
    for (int v = 0; v < 8; ++v) {
      float r = fast_sigm(xg0[v] + hg0[v]);
      float z = fast_sigm(xg1[v] + hg1[v]);
      float n = fast_tanh(xg2[v] + r * hg2[v]);
      hc[v] = n + z * (hc[v] - n); // == (1-z)*n + z*h
    }

    // ---- publish h to LDS first (next step's critical dependency) ----
#pragma unroll
    for (int v = 0; v < 8; ++v) {
      int m = half * 8 + v;
      hsh[m * 16 + l16] = hc[v];
    }

    // ---- write h_t to output (off critical path) ----
#pragma unroll
    for (int v = 0; v < 8; ++v)
      jb.out[ooff + v * jb.obs] = hc[v];

    // ---- advance / rotate double buffer ----
    xoff = xoff_n;
    ooff += doi;
#pragma unroll
    for (int c = 0; c < 8; ++c) xa[c] = xn[c];
  }

  if (jb.hT) {
#pragma unroll
    for (int v = 0; v < 8; ++v) {
      int m = half * 8 + v;
      jb.hT[(size_t)(row0 + m) * GRU_H + l16] = hc[v];
    }
  }
}

// ---------------------------------------------------------------------------
// Host orchestration
// ---------------------------------------------------------------------------

struct DirP { const float *whh, *wih, *bhh, *bih; int I; };

extern "C" void kernel_launch(void* const* d_in, const int* in_sizes, int n_in,
                              void* d_out, int out_size, void* d_ws, size_t ws_size,
                              hipStream_t stream) {
  (void)in_sizes; (void)n_in; (void)out_size; (void)ws_size;

  auto F = [&](int i) { return (const float*)d_in[i]; };
  const float* x = F(0); // (128, 2048, 32) batch-first
  const float* y = F(1); // (128, 2048, 25) batch-first
  float* out = (float*)d_out;
  float* ws  = (float*)d_ws;

  // --- params: jax pytree order (dict keys sorted; per-dir: Whh,Wih,bhh,bih) ---
  int idx = 2;
  auto take_b  = [&](int I) { DirP p; p.whh=F(idx++); p.wih=F(idx++);
                              p.bhh=F(idx++); p.bih=F(idx++); p.I=I; return p; };
  auto take_nb = [&](int I) { DirP p; p.whh=F(idx++); p.wih=F(idx++);
                              p.bhh=nullptr; p.bih=nullptr; p.I=I; return p; };

  DirP g00 = take_b(32);
  DirP g01[3]; for (int l = 0; l < 3; ++l) g01[l] = take_nb(16);
  DirP g02[4][2], g03[4][2], g04[4][2];
  for (int l = 0; l < 4; ++l) for (int d = 0; d < 2; ++d) g02[l][d] = take_b(l == 0 ? 16 : 32);
  for (int l = 0; l < 4; ++l) for (int d = 0; d < 2; ++d) g03[l][d] = take_b(l == 0 ? 16 : 32);
  for (int l = 0; l < 4; ++l) for (int d = 0; d < 2; ++d) g04[l][d] = take_b(l == 0 ? 16 : 32);
  DirP g10 = take_b(25);
  DirP g11[3]; for (int l = 0; l < 3; ++l) g11[l] = take_nb(16);
  DirP g12[4][2], g13[4][2], g14[4][2];
  for (int l = 0; l < 4; ++l) for (int d = 0; d < 2; ++d) g12[l][d] = take_b(l == 0 ? 16 : 32);
  for (int l = 0; l < 4; ++l) for (int d = 0; d < 2; ++d) g13[l][d] = take_b(l == 0 ? 16 : 32);
  for (int l = 0; l < 4; ++l) for (int d = 0; d < 2; ++d) g14[l][d] = take_b(l == 0 ? 16 : 32);

  // --- workspace layout (floats): ~192 MB ---
  const size_t S16 = (size_t)GRU_T * GRU_B * 16;
  const size_t S32 = S16 * 2;
  const size_t SH  = (size_t)8 * GRU_B * GRU_H;
  float* xs0 = ws;          float* xs1 = xs0 + S16;
  float* ys0 = xs1 + S16;   float* ys1 = ys0 + S16;
  float* xbA = ys1 + S16;   float* xbB = xbA + S32;
  float* ybA = xbB + S32;   float* ybB = ybA + S32;
  float* hx02 = ybB + S32;  float* hx03 = hx02 + SH;
  float* hy02 = hx03 + SH;  float* hy03 = hy02 + SH;

  const int TM16_t = GRU_B * 16, TM16_b = 16;   // time-major (T,B,16)
  const int TM32_t = GRU_B * 32, TM32_b = 32;   // time-major (T,B,32)
  const int BF_t = 32, BF_b = GRU_T * 32;       // batch-first (B,T,32)
  const size_t SLOT = (size_t)GRU_B * GRU_T * 32;

  auto mk = [](const float* in, int its, int ibs, const DirP& p, const float* h0,
               float* o, int ots, int obs, int ooff, float* hT, int rev) {
    ScanJob j;
    j.in = in; j.wih = p.wih; j.whh = p.whh; j.bih = p.bih; j.bhh = p.bhh;
    j.h0 = h0; j.out = o; j.hT = hT;
    j.I = p.I; j.its = its; j.ibs = ibs;
    j.ots = ots; j.obs = obs; j.ooff = ooff; j.rev = rev;
    return j;
  };
  auto launch = [&](std::initializer_list<ScanJob> js) {
    JobPack p{};
    int n = 0;
    for (const ScanJob& j : js) p.j[n++] = j;
    gru_scan<<<dim3(GRU_B / 16, n), dim3(32), 0, stream>>>(p);
  };

  // Stage A: first layers of both branches (batch-first inputs)
  launch({ mk(x, 32, GRU_T * 32, g00, nullptr, xs0, TM16_t, TM16_b, 0, nullptr, 0),
           mk(y, 25, GRU_T * 25, g10, nullptr, ys0, TM16_t, TM16_b, 0, nullptr, 0) });

  // Stage B: 3-layer unidirectional stacks (ping-pong xs0/xs1, ys0/ys1)
  {
    float* xi[3] = { xs0, xs1, xs0 }; float* xo[3] = { xs1, xs0, xs1 };
    float* yi[3] = { ys0, ys1, ys0 }; float* yo[3] = { ys1, ys0, ys1 };
    for (int l = 0; l < 3; ++l)
      launch({ mk(xi[l], TM16_t, TM16_b, g01[l], nullptr, xo[l], TM16_t, TM16_b, 0, nullptr, 0),
               mk(yi[l], TM16_t, TM16_b, g11[l], nullptr, yo[l], TM16_t, TM16_b, 0, nullptr, 0) });
  }

  // Stages C/D/E: 4-layer bidirectional stacks; fwd+rev of both branches per launch
  auto stack4 = [&](DirP (*gx)[2], DirP (*gy)[2],
                    const float* h0x, const float* h0y,
                    float* hTx, float* hTy,
                    float* foutx, float* fouty) {
    const float* xin[4]  = { xs1, xbA, xbB, xbA };
    float*       xout[4] = { xbA, xbB, xbA, foutx };
    const float* yin[4]  = { ys1, ybA, ybB, ybA };
    float*       yout[4] = { ybA, ybB, ybA, fouty };
    for (int l = 0; l < 4; ++l) {
      const int i_t = (l == 0) ? TM16_t : TM32_t;
      const int i_b = (l == 0) ? TM16_b : TM32_b;
      const int o_t = (l == 3) ? BF_t : TM32_t;
      const int o_b = (l == 3) ? BF_b : TM32_b;
      ScanJob js[4]; int n = 0;
      for (int d = 0; d < 2; ++d) {
        const size_t slot = (size_t)(l * 2 + d) * GRU_B * GRU_H;
        js[n++] = mk(xin[l], i_t, i_b, gx[l][d],
                     h0x ? h0x + slot : nullptr,
                     xout[l], o_t, o_b, d * 16,
                     hTx ? hTx + slot : nullptr, d);
      }
      for (int d = 0; d < 2; ++d) {
        const size_t slot = (size_t)(l * 2 + d) * GRU_B * GRU_H;
        js[n++] = mk(yin[l], i_t, i_b, gy[l][d],
                     h0y ? h0y + slot : nullptr,
                     yout[l], o_t, o_b, d * 16,
                     hTy ? hTy + slot : nullptr, d);
      }
      launch({ js[0], js[1], js[2], js[3] });
    }
  };

  stack4(g02, g12, nullptr, nullptr, hx02, hy02, out + 0 * SLOT, out + 3 * SLOT);
  stack4(g03, g13, hx02,    hy02,    hx03, hy03, out + 1 * SLOT, out + 4 * SLOT);
  stack4(g04, g14, hx03,    hy03,    nullptr, nullptr, out + 2 * SLOT, out + 5 * SLOT);
}